// AlphaKGNNStage_72387378806864
// MI455X (gfx1250) — compile-verified
//
#include <hip/hip_runtime.h>

#define N_NODES 100000
#define N_EDGES 800000
#define DIM     128
#define MAX_K   3
#define LDW     132   // padded LDS row stride (floats); 132*4=528 ≡ 0 (mod 16) so float4 chunks stay 16B aligned

typedef __attribute__((ext_vector_type(2))) float v2f;
typedef __attribute__((ext_vector_type(8))) float v8f;

#if defined(__gfx1250__) && __has_builtin(__builtin_amdgcn_global_load_async_to_lds_b128)
#define HAS_ASYNC_LDS 1
#else
#define HAS_ASYNC_LDS 0
#endif

// Builtin signature (from hipcc diagnostic): arg0 = v4i in addrspace(1) ("__device__"),
// arg1 = v4i in addrspace(3), arg2/arg3 = imm offset / cpol.
typedef int vint4 __attribute__((ext_vector_type(4)));
typedef __attribute__((address_space(1))) vint4 gas_vint4;
typedef __attribute__((address_space(3))) vint4 lds_vint4;

// 16-byte memory -> LDS copy; async (GLOBAL_LOAD_ASYNC_TO_LDS_B128, ASYNCcnt) when available.
__device__ __forceinline__ void copy16_to_lds(const float* __restrict__ g, float* lds) {
#if HAS_ASYNC_LDS
  __builtin_amdgcn_global_load_async_to_lds_b128((gas_vint4*)g, (lds_vint4*)lds, 0, 0);
#else
  *(float4*)lds = *(const float4*)g;
#endif
}

__device__ __forceinline__ void wait_async_lds() {
#if HAS_ASYNC_LDS
#if __has_builtin(__builtin_amdgcn_s_wait_asynccnt)
  __builtin_amdgcn_s_wait_asynccnt(0);
#else
  asm volatile("s_wait_asynccnt 0x0" ::: "memory");
#endif
#endif
}

// ---------- degree pipeline (layer-invariant) ----------
__global__ void deg_init_kernel(float* __restrict__ deg) {
  int i = blockIdx.x * blockDim.x + threadIdx.x;
  if (i < MAX_K * N_NODES) deg[i] = 1.0f;   // +1 self loop
}

__global__ void deg_count_kernel(const int* __restrict__ edge_index,
                                 const int* __restrict__ edge_attr,
                                 float* __restrict__ deg) {
  int e = blockIdx.x * blockDim.x + threadIdx.x;
  if (e >= N_EDGES) return;
  int dst = edge_index[N_EDGES + e];
  int k   = edge_attr[e];
  atomicAdd(&deg[k * N_NODES + dst], 1.0f);
}

// deg -> rsqrt(deg) in place; s[n] = sum_k a[k]/deg_k[n]; a3 = softmax(alpha)
__global__ void deg_finalize_kernel(float* __restrict__ deg,
                                    float* __restrict__ s,
                                    const float* __restrict__ alpha,
                                    float* __restrict__ a3) {
  int n = blockIdx.x * blockDim.x + threadIdx.x;
  if (n >= N_NODES) return;
  float m  = fmaxf(alpha[0], fmaxf(alpha[1], alpha[2]));
  float e0 = __expf(alpha[0] - m), e1 = __expf(alpha[1] - m), e2 = __expf(alpha[2] - m);
  float inv = 1.0f / (e0 + e1 + e2);
  float a[3] = { e0 * inv, e1 * inv, e2 * inv };
  if (n < MAX_K) a3[n] = a[n];
  float acc = 0.0f;
  for (int k = 0; k < MAX_K; ++k) {
    float d  = deg[k * N_NODES + n];
    float di = rsqrtf(d);
    deg[k * N_NODES + n] = di;      // becomes dinv
    acc += a[k] / d;
  }
  s[n] = acc;
}

// ---------- xw = x @ W[t] via V_WMMA_F32_16X16X4_F32 ----------
// 256 threads = 8 waves; block computes rows [32*blk, 32*blk+32) x 128 cols.
// W (64KB) + x tile (16KB) staged in LDS via async load-to-LDS (ASYNCcnt).
__global__ __launch_bounds__(256) void gemm_wmma_kernel(const float* __restrict__ x,
                                                        const float* __restrict__ Wt,
                                                        float* __restrict__ xw) {
  __shared__ float Ws[DIM * LDW];   // 128x128 W (padded)
  __shared__ float Xs[32 * LDW];    // 32x128 x tile
  const int tid  = threadIdx.x;
  const int base = blockIdx.x * 32;

  // W: 128 rows x 32 float4-chunks = 4096 chunks; 16 per thread.
  for (int i4 = tid; i4 < DIM * (DIM / 4); i4 += 256) {
    int r = i4 >> 5, c4 = i4 & 31;
    copy16_to_lds(Wt + (size_t)r * DIM + c4 * 4, &Ws[r * LDW + c4 * 4]);
  }
  // x tile: 32 rows x 32 chunks = 1024 chunks; 4 per thread.
  for (int i4 = tid; i4 < 32 * (DIM / 4); i4 += 256) {
    int r = i4 >> 5, c4 = i4 & 31;
    copy16_to_lds(x + (size_t)(base + r) * DIM + c4 * 4, &Xs[r * LDW + c4 * 4]);
  }
  wait_async_lds();
  __syncthreads();

  const int wave = tid >> 5, lane = tid & 31;
  const int rt   = wave >> 2;          // 0..1
  const int ct   = (wave & 3) * 2;     // 0,2,4,6
  const int half = lane >> 4;          // ISA: lanes 16-31 carry K+2 for A and B
  const int lm   = lane & 15;
  const int arow = rt * 16 + lm;

  v8f c0 = {}; v8f c1 = {};
  for (int kk = 0; kk < 32; ++kk) {            // K = 128 in steps of 4
    int kb = kk * 4 + half * 2;
    v2f A  = { Xs[arow * LDW + kb],              Xs[arow * LDW + kb + 1] };
    v2f B0 = { Ws[kb * LDW + ct * 16 + lm],      Ws[(kb + 1) * LDW + ct * 16 + lm] };
    v2f B1 = { Ws[kb * LDW + ct * 16 + 16 + lm], Ws[(kb + 1) * LDW + ct * 16 + 16 + lm] };
    c0 = __builtin_amdgcn_wmma_f32_16x16x4_f32(false, A, false, B0, (short)0, c0, false, false);
    c1 = __builtin_amdgcn_wmma_f32_16x16x4_f32(false, A, false, B1, (short)0, c1, false, false);
  }

  // C/D layout: VGPR j, lanes 0-15 -> M=j ; lanes 16-31 -> M=j+8 ; N = lane%16
  for (int j = 0; j < 8; ++j) {
    size_t orow = (size_t)(base + rt * 16 + half * 8 + j);
    xw[orow * DIM + ct * 16 + lm]      = c0[j];
    xw[orow * DIM + ct * 16 + 16 + lm] = c1[j];
  }
}

// ---------- zero scratch ----------
__global__ void zero_kernel(float* __restrict__ p, int n) {
  int i = blockIdx.x * blockDim.x + threadIdx.x;
  if (i < n) p[i] = 0.0f;
}

// ---------- edge scatter: one wave per edge, float4 per lane ----------
__global__ __launch_bounds__(256) void scatter_kernel(const int* __restrict__ edge_index,
                                                      const int* __restrict__ edge_attr,
                                                      const float* __restrict__ xw,
                                                      const float* __restrict__ dinv,
                                                      const float* __restrict__ a3,
                                                      float* __restrict__ agg) {
  int gt   = blockIdx.x * blockDim.x + threadIdx.x;
  int e    = gt >> 5;
  int lane = gt & 31;
  if (e >= N_EDGES) return;
  int s_ = edge_index[e];
  int d_ = edge_index[N_EDGES + e];
  int k  = edge_attr[e];
  float w = a3[k] * dinv[k * N_NODES + s_] * dinv[k * N_NODES + d_];
  float4 v = ((const float4*)(xw + (size_t)s_ * DIM))[lane];   // L2-resident gather
  float* ar = agg + (size_t)d_ * DIM + lane * 4;
  atomicAdd(ar + 0, w * v.x);
  atomicAdd(ar + 1, w * v.y);
  atomicAdd(ar + 2, w * v.z);
  atomicAdd(ar + 3, w * v.w);
}

// ---------- node update: residual + ReLU + L2 normalize (one wave per node) ----------
__global__ __launch_bounds__(256) void update_kernel(const float* __restrict__ xin,
                                                     const float* __restrict__ xw,
                                                     const float* __restrict__ agg,
                                                     const float* __restrict__ s,
                                                     const float* __restrict__ bt,
                                                     float* __restrict__ xout) {
  int gt   = blockIdx.x * blockDim.x + threadIdx.x;
  int n    = gt >> 5;
  int lane = gt & 31;
  if (n >= N_NODES) return;
  size_t off = (size_t)n * DIM + lane * 4;
  float4 xa = *(const float4*)(xin + off);
  float4 wv = *(const float4*)(xw  + off);
  float4 av = *(const float4*)(agg + off);
  float4 bv = *(const float4*)(bt + lane * 4);
  float sn = s[n];
  float4 v;
  v.x = xa.x + fmaxf(av.x + sn * wv.x + bv.x, 0.0f);
  v.y = xa.y + fmaxf(av.y + sn * wv.y + bv.y, 0.0f);
  v.z = xa.z + fmaxf(av.z + sn * wv.z + bv.z, 0.0f);
  v.w = xa.w + fmaxf(av.w + sn * wv.w + bv.w, 0.0f);
  float q = v.x * v.x + v.y * v.y + v.z * v.z + v.w * v.w;
  #pragma unroll
  for (int m = 16; m >= 1; m >>= 1) q += __shfl_xor(q, m, 32);  // wave32 reduction
  float scale = 1.0f / fmaxf(sqrtf(q), 1e-12f);
  float4 o = { v.x * scale, v.y * scale, v.z * scale, v.w * scale };
  *(float4*)(xout + off) = o;
}

extern "C" void kernel_launch(void* const* d_in, const int* in_sizes, int n_in,
                              void* d_out, int out_size, void* d_ws, size_t ws_size,
                              hipStream_t stream) {
  const float* x     = (const float*)d_in[0];
  const int*   ei    = (const int*)d_in[1];
  const int*   ea    = (const int*)d_in[2];
  const float* W     = (const float*)d_in[3];
  const float* b     = (const float*)d_in[4];
  const float* alpha = (const float*)d_in[5];
  float* out = (float*)d_out;

  char* ws = (char*)d_ws;
  float* xw   = (float*)ws; ws += sizeof(float) * (size_t)N_NODES * DIM;
  float* agg  = (float*)ws; ws += sizeof(float) * (size_t)N_NODES * DIM;
  float* dinv = (float*)ws; ws += sizeof(float) * (size_t)MAX_K * N_NODES;
  float* sarr = (float*)ws; ws += sizeof(float) * (size_t)N_NODES;
  float* a3   = (float*)ws;

  // layer-invariant degree / softmax precompute
  deg_init_kernel<<<(MAX_K * N_NODES + 255) / 256, 256, 0, stream>>>(dinv);
  deg_count_kernel<<<(N_EDGES + 255) / 256, 256, 0, stream>>>(ei, ea, dinv);
  deg_finalize_kernel<<<(N_NODES + 255) / 256, 256, 0, stream>>>(dinv, sarr, alpha, a3);

  for (int t = 0; t < 2; ++t) {
    const float* xin = (t == 0) ? x : out;
    gemm_wmma_kernel<<<N_NODES / 32, 256, 0, stream>>>(xin, W + (size_t)t * DIM * DIM, xw);
    zero_kernel<<<(N_NODES * DIM + 255) / 256, 256, 0, stream>>>(agg, N_NODES * DIM);
    scatter_kernel<<<((size_t)N_EDGES * 32 + 255) / 256, 256, 0, stream>>>(ei, ea, xw, dinv, a3, agg);
    update_kernel<<<((size_t)N_NODES * 32 + 255) / 256, 256, 0, stream>>>(xin, xw, agg, sarr,
                                                                          b + (size_t)t * DIM, out);
  }
}